// MassSpringSolver_60138132078746
// MI455X (gfx1250) — compile-verified
//
#include <hip/hip_runtime.h>
#include <hip/hip_bf16.h>

// Mass-spring implicit solver on gfx1250 (MI455X).
// CSR adjacency built on device -> pure-gather force & Laplacian matvec (no
// atomics in hot loops). 2-kernel fused CG iteration, f64 reductions via
// wave32 shuffles + global f64 atomics. State float4-padded for single-b128
// random gathers (working set ~20MB, L2-resident on 192MB L2). Adjacency slab
// per block staged to LDS with gfx1250 async-to-LDS ops when available.

#define NB 32
#define NV 8192
#define NE 65536
#define NSUB 8
#define CGIT 20
#define DTC 0.01f
#define KSP 10000.0f
#define DT2K 1.0f          /* DT*DT*K = 1e-4 * 1e4 */
#define EPSD 1e-12
#define SLAB_CAP 6144      /* ints; mean slab = 4096, sigma ~63 */

#if defined(__AMDGCN__) && __has_builtin(__builtin_amdgcn_global_load_async_to_lds_b32)
#define MSK_ASYNC_LDS 1
typedef __attribute__((address_space(1))) int as1_int;
typedef __attribute__((address_space(3))) int as3_int;
#else
#define MSK_ASYNC_LDS 0
#endif

__device__ __forceinline__ void msk_wait_async() {
#if defined(__AMDGCN__)
#if __has_builtin(__builtin_amdgcn_s_wait_asynccnt)
    __builtin_amdgcn_s_wait_asynccnt(0);
#else
    asm volatile("s_wait_asynccnt 0x0" ::: "memory");
#endif
#endif
}

// ---------------------------------------------------------------- CSR build
__global__ __launch_bounds__(256) void msk_count(const int* __restrict__ ei,
                                                 const int* __restrict__ ej,
                                                 int* __restrict__ counts) {
    int e = blockIdx.x * blockDim.x + threadIdx.x;
    if (e < NE) {
        atomicAdd(&counts[ei[e]], 1);
        atomicAdd(&counts[ej[e]], 1);
    }
}

__global__ __launch_bounds__(1024) void msk_scan(const int* __restrict__ counts,
                                                 int* __restrict__ row_ptr,
                                                 int* __restrict__ cursor) {
    __shared__ int lds[1024];
    int tid = threadIdx.x;
    int base = tid * 8;
    int c[8];
    int s = 0;
#pragma unroll
    for (int i = 0; i < 8; ++i) { int t = counts[base + i]; c[i] = s; s += t; }
    lds[tid] = s;
    __syncthreads();
    for (int off = 1; off < 1024; off <<= 1) {
        int t = (tid >= off) ? lds[tid - off] : 0;
        __syncthreads();
        lds[tid] += t;
        __syncthreads();
    }
    int chunk_off = lds[tid] - s;  // exclusive prefix of this chunk
#pragma unroll
    for (int i = 0; i < 8; ++i) {
        int v = chunk_off + c[i];
        row_ptr[base + i] = v;
        cursor[base + i]  = v;
    }
    if (tid == 1023) row_ptr[NV] = lds[1023];  // == 2*NE
}

__global__ __launch_bounds__(256) void msk_fill(const int* __restrict__ ei,
                                                const int* __restrict__ ej,
                                                int* __restrict__ cursor,
                                                int* __restrict__ adj_u,
                                                int* __restrict__ adj_e) {
    int e = blockIdx.x * blockDim.x + threadIdx.x;
    if (e < NE) {
        int a = ei[e], b = ej[e];
        int s0 = atomicAdd(&cursor[a], 1);
        adj_u[s0] = b; adj_e[s0] = e;
        int s1 = atomicAdd(&cursor[b], 1);
        adj_u[s1] = a; adj_e[s1] = e;
    }
}

// ---------------------------------------------------------------- reductions
__device__ __forceinline__ double wave_red(double v) {
#pragma unroll
    for (int off = 16; off > 0; off >>= 1) v += __shfl_down(v, off, 32);
    return v;
}

// Block-uniform (b, v) decomposition: 32 blocks per batch, b is an SGPR value.
__device__ __forceinline__ void msk_bv(int& b, int& v) {
    b = (int)(blockIdx.x >> 5);
    v = (((int)blockIdx.x & 31) << 8) + (int)threadIdx.x;
}

// ------------------------------------------------- force + CG init (substep)
__global__ __launch_bounds__(256) void msk_force_init(
        const float* __restrict__ pos_in, int ps,
        const float* __restrict__ vel_in, int vs,
        const float* __restrict__ action,
        const float* __restrict__ rest_len,
        const float* __restrict__ mass,
        const int* __restrict__ row_ptr,
        const int* __restrict__ adj_u,
        const int* __restrict__ adj_e,
        float4* __restrict__ x4, float4* __restrict__ r4, float4* __restrict__ p4,
        double* __restrict__ rs_out) {
    int b, v;
    msk_bv(b, v);
    const float* pb = pos_in + (size_t)b * NV * ps;
    float px = pb[v * ps + 0], py = pb[v * ps + 1], pz = pb[v * ps + 2];
    float m = mass[v];
    float fx = 0.f, fy = -9.8f * m, fz = 0.f;
    int kb = row_ptr[v], ke = row_ptr[v + 1];
    __builtin_prefetch(&adj_u[kb], 0, 0);
    __builtin_prefetch(&adj_e[kb], 0, 0);
    const float* act = action + (size_t)b * NE;
    for (int k = kb; k < ke; ++k) {
        int u = adj_u[k];
        int e = adj_e[k];
        float dx = px - pb[u * ps + 0];
        float dy = py - pb[u * ps + 1];
        float dz = pz - pb[u * ps + 2];
        float l = sqrtf(dx * dx + dy * dy + dz * dz);
        float rest = rest_len[e] * (1.0f + act[e]);
        float coef = -KSP * (l - rest) / fmaxf(l, 1e-6f);
        fx += coef * dx; fy += coef * dy; fz += coef * dz;
    }
    const float* vb = vel_in + (size_t)b * NV * vs + (size_t)v * vs;
    float bx = m * vb[0] + DTC * fx;
    float by = m * vb[1] + DTC * fy;
    float bz = m * vb[2] + DTC * fz;
    size_t i = (size_t)b * NV + v;
    x4[i] = make_float4(0.f, 0.f, 0.f, 0.f);   // safe: vel alias read above (same thread)
    float4 bb = make_float4(bx, by, bz, 0.f);
    r4[i] = bb;
    p4[i] = bb;
    double d = (double)bx * bx + (double)by * by + (double)bz * bz;
    __shared__ double red[8];
    d = wave_red(d);
    int lane = threadIdx.x & 31, wv = threadIdx.x >> 5;
    if (lane == 0) red[wv] = d;
    __syncthreads();
    if (threadIdx.x == 0) {
        double s = 0;
        for (int w = 0; w < 8; ++w) s += red[w];
        atomicAdd(&rs_out[b], s);
    }
}

// --------------------------------------------------- matvec + 3 dot products
__global__ __launch_bounds__(256) void msk_matvec(
        const float4* __restrict__ p4, const float4* __restrict__ r4,
        const float* __restrict__ mass,
        const int* __restrict__ row_ptr, const int* __restrict__ adj_u,
        float4* __restrict__ Ap4, double* __restrict__ dots) {
    __shared__ int slab[SLAB_CAP];
    __shared__ double red[24];
    int b, v;
    msk_bv(b, v);
    size_t base = (size_t)b * NV;

    // Stage this block's contiguous adjacency slab into LDS (scalar bounds).
    int v0 = (((int)blockIdx.x & 31) << 8);
    int kb_blk = row_ptr[v0];
    int slabLen = row_ptr[v0 + 256] - kb_blk;
    bool useLds = (slabLen <= SLAB_CAP);
    if (useLds) {
        for (int idx = threadIdx.x; idx < slabLen; idx += 256) {
#if MSK_ASYNC_LDS
            __builtin_amdgcn_global_load_async_to_lds_b32(
                (as1_int*)(uintptr_t)(adj_u + kb_blk + idx),
                (as3_int*)(slab + idx), 0, 0);
#else
            slab[idx] = adj_u[kb_blk + idx];
#endif
        }
#if MSK_ASYNC_LDS
        msk_wait_async();
#endif
        __syncthreads();
    }

    float4 pv = p4[base + v];
    int kb = row_ptr[v], ke = row_ptr[v + 1];
    float ax = 0.f, ay = 0.f, az = 0.f;
    if (useLds) {
        int lo = kb - kb_blk, hi = ke - kb_blk;
        for (int k = lo; k < hi; ++k) {
            int u = slab[k];
            float4 pu = p4[base + u];    // single b128 gather, L2-resident
            ax += pu.x; ay += pu.y; az += pu.z;
        }
    } else {
        for (int k = kb; k < ke; ++k) {
            int u = adj_u[k];
            float4 pu = p4[base + u];
            ax += pu.x; ay += pu.y; az += pu.z;
        }
    }
    float deg = (float)(ke - kb);
    float m = mass[v];
    float apx = m * pv.x + DT2K * (deg * pv.x - ax);
    float apy = m * pv.y + DT2K * (deg * pv.y - ay);
    float apz = m * pv.z + DT2K * (deg * pv.z - az);
    Ap4[base + v] = make_float4(apx, apy, apz, 0.f);
    float4 rv = r4[base + v];
    double pAp  = (double)pv.x * apx + (double)pv.y * apy + (double)pv.z * apz;
    double rAp  = (double)rv.x * apx + (double)rv.y * apy + (double)rv.z * apz;
    double ApAp = (double)apx * apx + (double)apy * apy + (double)apz * apz;
    pAp = wave_red(pAp); rAp = wave_red(rAp); ApAp = wave_red(ApAp);
    int lane = threadIdx.x & 31, wv = threadIdx.x >> 5;
    if (lane == 0) { red[wv] = pAp; red[8 + wv] = rAp; red[16 + wv] = ApAp; }
    __syncthreads();
    if (threadIdx.x == 0) {
        double s0 = 0, s1 = 0, s2 = 0;
        for (int w = 0; w < 8; ++w) { s0 += red[w]; s1 += red[8 + w]; s2 += red[16 + w]; }
        atomicAdd(&dots[b * 3 + 0], s0);
        atomicAdd(&dots[b * 3 + 1], s1);
        atomicAdd(&dots[b * 3 + 2], s2);
    }
}

// ----------------------- fused CG update: alpha/beta, x/r/p AXPYs, rs, finalize
__global__ __launch_bounds__(256) void msk_update(
        float4* __restrict__ x4, float4* __restrict__ r4, float4* __restrict__ p4,
        const float4* __restrict__ Ap4,
        const double* __restrict__ rs_in, const double* __restrict__ dots,
        double* __restrict__ rs_out, int last,
        const float* __restrict__ pos_in, int ps,
        float4* __restrict__ pos_out4, float* __restrict__ traj, int sstep) {
    int b, v;
    msk_bv(b, v);
    size_t i = (size_t)b * NV + v;
    // b is SGPR-uniform -> these become scalar loads.
    double rs   = rs_in[b];
    double pAp  = dots[b * 3 + 0];
    double rAp  = dots[b * 3 + 1];
    double ApAp = dots[b * 3 + 2];
    double alpha = rs / (pAp + EPSD);
    double rs_exp = rs - 2.0 * alpha * rAp + alpha * alpha * ApAp;
    if (rs_exp < 0.0) rs_exp = 0.0;
    double beta = rs_exp / (rs + EPSD);
    float af = (float)alpha, bf = (float)beta;

    float4 xv = x4[i], rv = r4[i], pv = p4[i], av = Ap4[i];
    float nxx = xv.x + af * pv.x;
    float nxy = xv.y + af * pv.y;
    float nxz = xv.z + af * pv.z;
    x4[i] = make_float4(nxx, nxy, nxz, 0.f);
    float rx = rv.x - af * av.x;
    float ry = rv.y - af * av.y;
    float rz = rv.z - af * av.z;
    r4[i] = make_float4(rx, ry, rz, 0.f);
    double rr = (double)rx * rx + (double)ry * ry + (double)rz * rz;
    if (!last) {
        p4[i] = make_float4(rx + bf * pv.x, ry + bf * pv.y, rz + bf * pv.z, 0.f);
    } else {
        // pos_new = pos + DT * v_new ; store into state + trajectory output
        const float* pb = pos_in + (size_t)b * NV * ps + (size_t)v * ps;
        float ppx = pb[0] + DTC * nxx;
        float ppy = pb[1] + DTC * nxy;
        float ppz = pb[2] + DTC * nxz;
        pos_out4[i] = make_float4(ppx, ppy, ppz, 0.f);
        size_t o = (((size_t)b * NSUB + sstep) * NV + (size_t)v) * 3;
        traj[o + 0] = ppx;
        traj[o + 1] = ppy;
        traj[o + 2] = ppz;
    }
    // direct r_new.r_new reduction -> next iteration's alpha numerator
    __shared__ double red[8];
    rr = wave_red(rr);
    int lane = threadIdx.x & 31, wv = threadIdx.x >> 5;
    if (lane == 0) red[wv] = rr;
    __syncthreads();
    if (threadIdx.x == 0) {
        double s = 0;
        for (int w = 0; w < 8; ++w) s += red[w];
        atomicAdd(&rs_out[b], s);
    }
}

// ---------------------------------------------------------------- launcher
extern "C" void kernel_launch(void* const* d_in, const int* in_sizes, int n_in,
                              void* d_out, int out_size, void* d_ws, size_t ws_size,
                              hipStream_t stream) {
    const float* action   = (const float*)d_in[0];  // (B, NE)
    const float* pos0     = (const float*)d_in[1];  // (B, NV, 3)
    const float* vel0     = (const float*)d_in[2];  // (B, NV, 3)
    const float* rest_len = (const float*)d_in[3];  // (NE,)
    const float* mass     = (const float*)d_in[4];  // (NV,)
    const int*   ei       = (const int*)d_in[5];    // (NE,)
    const int*   ej       = (const int*)d_in[6];    // (NE,)
    float* traj = (float*)d_out;                    // (B, NSUB, NV, 3)

    char* w = (char*)d_ws;
    auto carve = [&](size_t bytes) -> void* {
        void* ptr = (void*)w;
        w += (bytes + 255) & ~(size_t)255;
        return ptr;
    };
    int*    counts  = (int*)carve((size_t)NV * 4);
    int*    row_ptr = (int*)carve((size_t)(NV + 1) * 4);
    int*    cursor  = (int*)carve((size_t)NV * 4);
    int*    adj_u   = (int*)carve((size_t)2 * NE * 4);
    int*    adj_e   = (int*)carve((size_t)2 * NE * 4);
    float4* pos4    = (float4*)carve((size_t)NB * NV * 16);
    float4* x4      = (float4*)carve((size_t)NB * NV * 16);   // doubles as vel
    float4* r4      = (float4*)carve((size_t)NB * NV * 16);
    float4* p4      = (float4*)carve((size_t)NB * NV * 16);
    float4* Ap4     = (float4*)carve((size_t)NB * NV * 16);
    double* rs_sl   = (double*)carve((size_t)NSUB * (CGIT + 1) * NB * 8);
    double* dot_sl  = (double*)carve((size_t)NSUB * CGIT * NB * 3 * 8);

    (void)hipMemsetAsync(counts, 0, (size_t)NV * 4, stream);
    (void)hipMemsetAsync(rs_sl, 0, (size_t)NSUB * (CGIT + 1) * NB * 8, stream);
    (void)hipMemsetAsync(dot_sl, 0, (size_t)NSUB * CGIT * NB * 3 * 8, stream);

    msk_count<<<NE / 256, 256, 0, stream>>>(ei, ej, counts);
    msk_scan<<<1, 1024, 0, stream>>>(counts, row_ptr, cursor);
    msk_fill<<<NE / 256, 256, 0, stream>>>(ei, ej, cursor, adj_u, adj_e);

    const int grid = (NB * NV) / 256;  // 1024 blocks, 32 per batch
    for (int s = 0; s < NSUB; ++s) {
        const float* pin = (s == 0) ? pos0 : (const float*)pos4;
        int ps = (s == 0) ? 3 : 4;
        const float* vin = (s == 0) ? vel0 : (const float*)x4;
        int vs = (s == 0) ? 3 : 4;
        double* rs_base = rs_sl + (size_t)s * (CGIT + 1) * NB;
        msk_force_init<<<grid, 256, 0, stream>>>(pin, ps, vin, vs, action, rest_len,
                                                 mass, row_ptr, adj_u, adj_e,
                                                 x4, r4, p4, rs_base);
        for (int t = 0; t < CGIT; ++t) {
            double* dots = dot_sl + ((size_t)s * CGIT + t) * NB * 3;
            msk_matvec<<<grid, 256, 0, stream>>>(p4, r4, mass, row_ptr, adj_u, Ap4, dots);
            msk_update<<<grid, 256, 0, stream>>>(x4, r4, p4, Ap4,
                                                 rs_base + (size_t)t * NB, dots,
                                                 rs_base + (size_t)(t + 1) * NB,
                                                 (t == CGIT - 1) ? 1 : 0,
                                                 pin, ps, pos4, traj, s);
        }
    }
}